// TenHotEncodeLayer_53566832115799
// MI455X (gfx1250) — compile-verified
//
#include <hip/hip_runtime.h>
#include <hip/hip_bf16.h>

// Ten-hot encode: out[B, 32000] = 0, then out[n, x[n,j]] = 1.0 (set semantics).
// Pure store-bandwidth problem: 1.024 GB of output vs 23.3 TB/s HBM => ~44us
// floor. Strategy: streaming non-temporal b128 zero-fill (output >> 192MB L2,
// never re-read, so NT avoids L2 rinse), then a tiny scatter kernel for the
// 81,920 ones. Stream-serialized kernels guarantee zero-then-set ordering.
//
// v2: hoist bounds checking out of the store loop. The grid exactly tiles the
// output (32,000 blocks x 2048 chunks), so the common case is one uniform
// scalar branch per block followed by 8 unconditional b128 NT stores with
// immediate offsets (no per-store v_cmpx/exec churn, zeros materialized once).

typedef float v4f __attribute__((ext_vector_type(4)));

#define NUM_TOKENS 32000
#define KHOT 10

#define ZERO_THREADS 256
#define ZERO_UNROLL  8   // 8 x b128 NT stores per thread, coalesced per wave
#define CHUNKS_PER_BLOCK (ZERO_THREADS * ZERO_UNROLL)  // 2048

__global__ __launch_bounds__(ZERO_THREADS)
void tenhot_zero_kernel(float* __restrict__ out, long long nchunks) {
    const long long blockBase = (long long)blockIdx.x * CHUNKS_PER_BLOCK;
    v4f* __restrict__ p = (v4f*)out + blockBase + threadIdx.x;
    v4f z = (v4f)0.0f;

    if (blockBase + CHUNKS_PER_BLOCK <= nchunks) {
        // Fast path (always taken when grid tiles exactly): straight-line
        // clause of 8 coalesced global_store_b128 th:TH_STORE_NT.
        // Lanes store consecutive 16B chunks -> 512B contiguous per wave32
        // store instruction.
#pragma unroll
        for (int j = 0; j < ZERO_UNROLL; ++j) {
            __builtin_nontemporal_store(z, p + j * ZERO_THREADS);
        }
    } else {
        // Generic tail path (unused at this problem size).
        const long long base = blockBase + threadIdx.x;
#pragma unroll
        for (int j = 0; j < ZERO_UNROLL; ++j) {
            if (base + (long long)j * ZERO_THREADS < nchunks) {
                __builtin_nontemporal_store(z, p + j * ZERO_THREADS);
            }
        }
    }
}

__global__ __launch_bounds__(256)
void tenhot_scatter_kernel(const int* __restrict__ x,
                           float* __restrict__ out, int n) {
    int i = blockIdx.x * blockDim.x + threadIdx.x;
    if (i >= n) return;
    int row = i / KHOT;                 // strength-reduced by compiler
    unsigned idx = (unsigned)x[i];      // unsigned compare drops negatives too
    if (idx < (unsigned)NUM_TOKENS) {
        out[(long long)row * NUM_TOKENS + (long long)idx] = 1.0f;
    }
}

extern "C" void kernel_launch(void* const* d_in, const int* in_sizes, int n_in,
                              void* d_out, int out_size, void* d_ws, size_t ws_size,
                              hipStream_t stream) {
    const int* x = (const int*)d_in[0];   // integer input arrives as int32
    float* out = (float*)d_out;

    // ---- Pass 1: streaming NT zero-fill of the whole output ----
    long long nfloats = (long long)out_size;          // 262,144,000
    long long nchunks = nfloats >> 2;                 // float4 chunks (16B-aligned hipMalloc)
    int zero_blocks = (int)((nchunks + CHUNKS_PER_BLOCK - 1) / CHUNKS_PER_BLOCK); // 32,000
    tenhot_zero_kernel<<<zero_blocks, ZERO_THREADS, 0, stream>>>(out, nchunks);

    // ---- Pass 2: scatter the ones (81,920 threads; negligible traffic) ----
    int n = in_sizes[0];                              // BATCH * KHOT = 81,920
    int sc_blocks = (n + 255) / 256;
    tenhot_scatter_kernel<<<sc_blocks, 256, 0, stream>>>(x, out, n);
}